// DummyMoveNet_30880814858791
// MI455X (gfx1250) — compile-verified
//
#include <hip/hip_runtime.h>
#include <stdint.h>

// MoveNet decode for MI455X (gfx1250).
// Bandwidth-bound: optimal traffic ~85 MB (hm + ct + tiny gathers) vs 405 MB naive.
// Uses the CDNA5 Tensor Data Mover to DMA 96x96 f32 channels into LDS
// (TENSORcnt + s_wait_tensorcnt), wave32 shfl argmax reductions.

#define HT 48
#define WT 48
#define HS 96
#define WS 96
#define NJ 17
#define NB 128

typedef __attribute__((ext_vector_type(4))) unsigned int v4u;
typedef __attribute__((ext_vector_type(8))) int          v8i;
typedef __attribute__((ext_vector_type(4))) int          v4i;

#if __has_builtin(__builtin_amdgcn_tensor_load_to_lds) && __has_builtin(__builtin_amdgcn_s_wait_tensorcnt)
#define HAVE_TDM 1
#else
#define HAVE_TDM 0
#endif

// DMA one 96x96 f32 channel (36 KB) from global into LDS via the TDM.
// Only wave 0 issues it (TDM ignores EXEC -> must be behind a real branch),
// waits TENSORcnt==0, then the whole block syncs.
__device__ __forceinline__ void load_tile_96x96(float* lds_tile, const float* gsrc, int tid) {
#if HAVE_TDM
  if (tid < 32) {  // wave 0 only
    // Low 32 bits of the flat shared-aperture pointer == wave-relative LDS byte offset.
    uint32_t lds_addr = (uint32_t)(uintptr_t)lds_tile;
    uint64_t ga = (uint64_t)(uintptr_t)gsrc;
    // D# group 0: count=1 (valid), lds_addr[63:32], global_addr[120:64], type=2 [127:126]
    v4u g0;
    g0.x = 0x1u;
    g0.y = lds_addr;
    g0.z = (uint32_t)(ga & 0xffffffffu);
    g0.w = (uint32_t)((ga >> 32) & 0x01ffffffu) | (2u << 30);
    // D# group 1: data_size=4B; tensor_dim0/1 = 96; tile_dim0/1 = 96; dim0 stride = 96;
    // dim1 stride = 9216 (unused: tile_dim2 = 0).
    v8i g1;
    g1[0] = (int)(2u << 16);                     // data_size=2 -> 4 bytes
    g1[1] = (int)((uint32_t)WS << 16);           // tensor_dim0[15:0] @ bits[63:48]
    g1[2] = (int)((uint32_t)HS << 16);           // tensor_dim1[15:0] @ bits[95:80]
    g1[3] = (int)((uint32_t)WS << 16);           // tile_dim0 @ bits[127:112]
    g1[4] = HS;                                  // tile_dim1 @ [143:128], tile_dim2=0
    g1[5] = WS;                                  // tensor_dim0_stride[31:0]
    g1[6] = (int)((uint32_t)(HS * WS) << 16);    // tensor_dim1_stride[15:0] @ [223:208]
    g1[7] = 0;
    v4i g2 = {0, 0, 0, 0};
    v4i g3 = {0, 0, 0, 0};
#if defined(__clang_major__) && (__clang_major__ >= 23)
    v8i gz = {0, 0, 0, 0, 0, 0, 0, 0};
    __builtin_amdgcn_tensor_load_to_lds(g0, g1, g2, g3, gz, 0);
#else
    __builtin_amdgcn_tensor_load_to_lds(g0, g1, g2, g3, 0);
#endif
    __builtin_amdgcn_s_wait_tensorcnt(0);
  }
#else
  // Fallback: cooperative vectorized copy (global_load_b128 -> ds_store_b128).
  for (int i = tid; i < (HS * WS) / 4; i += 256) {
    ((float4*)lds_tile)[i] = ((const float4*)gsrc)[i];
  }
#endif
  __syncthreads();
}

// Bilinear 96->48 with align_corners=False is exactly a 2x2 average (separable form
// to match jax rounding order as closely as possible). p points at (2y, 2x), stride WS.
__device__ __forceinline__ float avg4(const float* p) {
  return 0.5f * (0.5f * (p[0] + p[1]) + 0.5f * (p[WS] + p[WS + 1]));
}

// argmax combine with jnp.argmax tie-breaking (first/lowest index wins).
__device__ __forceinline__ void amax2(float& v, int& i, float ov, int oi) {
  if (ov > v || (ov == v && oi < i)) { v = ov; i = oi; }
}

// Block-wide argmax over 256 threads (8 wave32). Result valid in thread 0.
__device__ __forceinline__ void block_argmax(float& v, int& i, float* s_v, int* s_i, int tid) {
  for (int m = 16; m > 0; m >>= 1) {
    float ov = __shfl_xor(v, m, 32);
    int   oi = __shfl_xor(i, m, 32);
    amax2(v, i, ov, oi);
  }
  if ((tid & 31) == 0) { s_v[tid >> 5] = v; s_i[tid >> 5] = i; }
  __syncthreads();
  if (tid == 0) {
    for (int w = 1; w < 8; ++w) amax2(v, i, s_v[w], s_i[w]);
  }
}

// Pass 1: per-batch center argmax over resized ct, then gather reg at the center
// pixel (only 4 input texels per channel -- avoids resizing rg at all).
__global__ __launch_bounds__(256) void movenet_center(const float* __restrict__ ct,
                                                      const float* __restrict__ rg,
                                                      float* __restrict__ ws_reg) {
  __shared__ float tile[HS * WS];
  __shared__ float s_v[8];
  __shared__ int   s_i[8];
  __shared__ int   s_c[2];
  const int b = blockIdx.x;
  const int tid = threadIdx.x;

  load_tile_96x96(tile, ct + (size_t)b * (HS * WS), tid);

  float best = -__builtin_inff();
  int   bi = 0;
  for (int p = tid; p < HT * WT; p += 256) {   // increasing p => first-index-wins per thread
    int y = p / WT, x = p - y * WT;
    float v = avg4(&tile[(2 * y) * WS + 2 * x]);
    if (v > best) { best = v; bi = p; }
  }
  block_argmax(best, bi, s_v, s_i, tid);
  if (tid == 0) { s_c[0] = bi / WT; s_c[1] = bi - (bi / WT) * WT; }
  __syncthreads();

  if (tid < 2 * NJ) {
    const int cy = s_c[0], cx = s_c[1];
    const int c = tid, j = c >> 1, comp = c & 1;   // channel 2j -> x, 2j+1 -> y
    float v = avg4(rg + ((size_t)b * 2 * NJ + c) * (HS * WS) + (size_t)(2 * cy) * WS + 2 * cx);
    float base = comp ? (float)cy : (float)cx;
    float coord = fminf(fmaxf(base + v + 0.5f, 0.0f), (float)(WT - 1));
    ws_reg[comp * (NB * NJ) + b * NJ + j] = coord;
  }
}

// Pass 2: per (b, joint) distance-weighted argmax over resized hm, then gather
// offsets (4 texels x 2 channels) and score, write x,y,s.
__global__ __launch_bounds__(256) void movenet_joint(const float* __restrict__ hm,
                                                     const float* __restrict__ of,
                                                     const float* __restrict__ ws_reg,
                                                     float* __restrict__ out) {
  __shared__ float tile[HS * WS];
  __shared__ float s_v[8];
  __shared__ int   s_i[8];
  const int bj = blockIdx.x;            // b*NJ + j
  const int b = bj / NJ, j = bj - b * NJ;
  const int tid = threadIdx.x;

  const float regx = ws_reg[bj];
  const float regy = ws_reg[NB * NJ + bj];

  load_tile_96x96(tile, hm + (size_t)bj * (HS * WS), tid);

  float best = -__builtin_inff();
  int   bi = 0;
  for (int p = tid; p < HT * WT; p += 256) {
    int y = p / WT, x = p - y * WT;
    float v = avg4(&tile[(2 * y) * WS + 2 * x]);
    float dx = (float)x - regx;
    float dy = (float)y - regy;
    float t = v / __builtin_sqrtf(dx * dx + dy * dy + 1e-9f) / 1.8f;
    if (t > best) { best = t; bi = p; }
  }
  block_argmax(best, bi, s_v, s_i, tid);

  if (tid == 0) {
    const int jy = bi / WT, jx = bi - (bi / WT) * WT;
    const float* p0 = of + ((size_t)b * 2 * NJ + 2 * j) * (HS * WS) + (size_t)(2 * jy) * WS + 2 * jx;
    float offx  = avg4(p0);
    float offy  = avg4(p0 + HS * WS);
    float score = avg4(&tile[(2 * jy) * WS + 2 * jx]);
    float xn = ((float)jx + offx) / (float)WT;
    float yn = ((float)jy + offy) / (float)HT;
    float* o = out + (size_t)b * (3 * NJ) + j * 3;
    o[0] = xn;
    o[1] = yn;
    o[2] = score;
  }
}

extern "C" void kernel_launch(void* const* d_in, const int* in_sizes, int n_in,
                              void* d_out, int out_size, void* d_ws, size_t ws_size,
                              hipStream_t stream) {
  (void)in_sizes; (void)n_in; (void)out_size; (void)ws_size;
  const float* hm = (const float*)d_in[0];
  const float* ct = (const float*)d_in[1];
  const float* rg = (const float*)d_in[2];
  const float* of = (const float*)d_in[3];
  float* ws_reg = (float*)d_ws;  // [2][NB*NJ] floats: reg_x then reg_y (17 KB)

  movenet_center<<<NB, 256, 0, stream>>>(ct, rg, ws_reg);
  movenet_joint<<<NB * NJ, 256, 0, stream>>>(hm, of, ws_reg, (float*)d_out);
}